// DiscreteContinuousConvS2_46402826666240
// MI455X (gfx1250) — compile-verified
//
#include <hip/hip_runtime.h>

// Problem constants (match reference config)
#define K_KER    9
#define NLAT_OUT 181
#define NLON_OUT 360
#define NLAT_IN  361
#define NLON_IN  720
#define C_IN     32
#define C_OUT    32
#define NNZ_     120000
#define CK       (C_IN * K_KER)            // 288
#define NTOT     (NLAT_OUT * NLON_OUT)     // 65160
#define TILE_EL  (C_IN * NLON_OUT)         // 11520 (46 KB LDS tile)
#define LIST_CAP 4096

typedef __attribute__((ext_vector_type(2))) float v2f;
typedef __attribute__((ext_vector_type(8))) float v8f;

// ---------------------------------------------------------------------------
// Stage 1: one workgroup per (k, ho) segment. Deterministic ballot compaction
// of matching nnz entries, then cooperative LDS accumulation (no atomics).
// Writes xk laid out [ck = c*9+k][n = ho*360+wo] for coalesced WMMA B loads.
// ---------------------------------------------------------------------------
__global__ __launch_bounds__(256) void disco_scatter_kernel(
    const float* __restrict__ x, const int* __restrict__ ker_idx,
    const int* __restrict__ row_idx, const int* __restrict__ col_idx,
    const float* __restrict__ vals, float* __restrict__ xk) {
  __shared__ float tile[TILE_EL];
  __shared__ int   list[LIST_CAP];
  __shared__ int   wcount[8];
  __shared__ int   chunk_base;

  const int t    = threadIdx.x;
  const int lane = t & 31;
  const int wave = t >> 5;
  const int ho   = blockIdx.x;   // 0..180
  const int k    = blockIdx.y;   // 0..8

  if (t == 0) chunk_base = 0;
  for (int p = t; p < TILE_EL; p += 256) tile[p] = 0.0f;
  __syncthreads();

  // --- Phase A: deterministic compaction (list kept in ascending-e order) ---
  for (int base = 0; base < NNZ_; base += 256) {
    const int e = base + t;
    const bool m = (e < NNZ_) && (ker_idx[e] == k) && (row_idx[e] == ho);
    const unsigned mask = (unsigned)__ballot(m);
    const int wprefix = __popc(mask & ((1u << lane) - 1u));
    if (lane == 0) wcount[wave] = __popc(mask);
    __syncthreads();  // wcount ready; chunk_base from previous chunk stable
    int woff = 0, stot = 0;
#pragma unroll
    for (int wv = 0; wv < 8; ++wv) {
      const int c = wcount[wv];
      stot += c;
      if (wv < wave) woff += c;
    }
    if (m) {
      const int pos = chunk_base + woff + wprefix;
      if (pos < LIST_CAP) list[pos] = e;
    }
    __syncthreads();  // all reads of chunk_base/wcount complete
    if (t == 0) chunk_base += stot;
  }
  __syncthreads();
  int nmatch = chunk_base;
  if (nmatch > LIST_CAP) nmatch = LIST_CAP;

  // --- Phase B: cooperative accumulation; thread t owns tile[p], p ≡ t (256)
  for (int i = 0; i < nmatch; ++i) {
    const int   e  = list[i];
    const int   ci = col_idx[e];
    const int   hi = ci / NLON_IN;
    const int   wi = ci - hi * NLON_IN;
    const float v  = vals[e];
    const float* __restrict__ xrow = x + hi * NLON_IN;
    for (int p = t; p < TILE_EL; p += 256) {
      const int c  = p / NLON_OUT;
      const int wo = p - c * NLON_OUT;
      int w2 = wi + 2 * wo;
      if (w2 >= NLON_IN) w2 -= NLON_IN;
      tile[p] += v * xrow[c * (NLAT_IN * NLON_IN) + w2];
    }
  }
  // Same thread wrote tile[p] it now reads: no barrier needed.

  // --- Phase C: writeback, fully initializes this segment's xk rows ---
  const int nbase = ho * NLON_OUT;
  for (int p = t; p < TILE_EL; p += 256) {
    const int c  = p / NLON_OUT;
    const int wo = p - c * NLON_OUT;
    xk[(size_t)(c * K_KER + k) * NTOT + nbase + wo] = tile[p];
  }
}

// ---------------------------------------------------------------------------
// Stage 2: out[32, 65160] = W[32, 288] x xk[288, 65160] + bias, via
// V_WMMA_F32_16X16X4_F32. One 16x16 D tile per wave; 72 K-steps in two
// accumulator chains. Column clamp keeps all B loads in-bounds.
// ---------------------------------------------------------------------------
__global__ __launch_bounds__(256) void disco_gemm_wmma_kernel(
    const float* __restrict__ xk, const float* __restrict__ w,
    const float* __restrict__ bias, float* __restrict__ out) {
  const int lane  = threadIdx.x & 31;
  const int wave  = threadIdx.x >> 5;
  const int ntile = blockIdx.x * 8 + wave;
  const int num_ntiles = (NTOT + 15) / 16;  // 4073
  if (ntile >= num_ntiles) return;          // wave-uniform: EXEC stays all-1s
  const int m0   = blockIdx.y * 16;         // 0 or 16
  const int half = lane >> 4;               // 0: K=0,1  1: K=2,3 half-wave
  const int l15  = lane & 15;
  const int n    = ntile * 16 + l15;
  const int nb   = (n < NTOT) ? n : (NTOT - 1);  // clamp (garbage cols dropped)

  const float* __restrict__ wrow = w + (m0 + l15) * CK;  // A row M = m0+l15
  v8f acc0 = {};
  v8f acc1 = {};
#pragma unroll 4
  for (int k0 = 0; k0 < CK; k0 += 8) {
    {
      const int ka = k0 + half * 2;
      v2f a, b;
      a.x = wrow[ka];
      a.y = wrow[ka + 1];
      b.x = xk[(size_t)ka * NTOT + nb];
      b.y = xk[(size_t)(ka + 1) * NTOT + nb];
      acc0 = __builtin_amdgcn_wmma_f32_16x16x4_f32(false, a, false, b,
                                                   (short)0, acc0, false, false);
    }
    {
      const int ka = k0 + 4 + half * 2;
      v2f a, b;
      a.x = wrow[ka];
      a.y = wrow[ka + 1];
      b.x = xk[(size_t)ka * NTOT + nb];
      b.y = xk[(size_t)(ka + 1) * NTOT + nb];
      acc1 = __builtin_amdgcn_wmma_f32_16x16x4_f32(false, a, false, b,
                                                   (short)0, acc1, false, false);
    }
  }
  const v8f acc = acc0 + acc1;

  if (n < NTOT) {
#pragma unroll
    for (int r = 0; r < 8; ++r) {
      const int o = m0 + r + half * 8;  // D VGPR r: lanes 0-15 -> M=r, 16-31 -> M=r+8
      out[(size_t)o * NTOT + n] = acc[r] + bias[o];
    }
  }
}

// ---------------------------------------------------------------------------
extern "C" void kernel_launch(void* const* d_in, const int* in_sizes, int n_in,
                              void* d_out, int out_size, void* d_ws, size_t ws_size,
                              hipStream_t stream) {
  const float* x     = (const float*)d_in[0];  // [1,32,361,720] f32
  const int*   ker   = (const int*)d_in[1];    // [120000] i32
  const int*   row   = (const int*)d_in[2];    // [120000] i32
  const int*   col   = (const int*)d_in[3];    // [120000] i32
  const float* vals  = (const float*)d_in[4];  // [120000] f32
  const float* wgt   = (const float*)d_in[5];  // [32,32,9] f32
  const float* bias  = (const float*)d_in[6];  // [32] f32
  float*       out   = (float*)d_out;          // [1,32,181,360] f32
  float*       xk    = (float*)d_ws;           // needs 288*65160*4 = 75,064,320 B

  dim3 g1(NLAT_OUT, K_KER);                    // (181, 9) segments
  disco_scatter_kernel<<<g1, 256, 0, stream>>>(x, ker, row, col, vals, xk);

  const int num_ntiles = (NTOT + 15) / 16;     // 4073
  dim3 g2((num_ntiles + 7) / 8, C_OUT / 16);   // (510, 2)
  disco_gemm_wmma_kernel<<<g2, 256, 0, stream>>>(xk, wgt, bias, out);
}